// NewtonImplicitNet_62543313764803
// MI455X (gfx1250) — compile-verified
//
#include <hip/hip_runtime.h>
#include <math.h>
#include <stdint.h>

typedef __attribute__((ext_vector_type(2))) float v2f;
typedef __attribute__((ext_vector_type(8))) float v8f;

#define B_SZ    2048
#define U_SZ    128
#define DIN_SZ  784
#define DOUT_SZ 10
#define MAXIT   50

// ---------------------------------------------------------------------------
// C[M,N] = A[M,K] @ W[N,K]^T + bias   (fp32 WMMA 16x16x4)
// bias from bias_vec[N] or bias_mat[M,N] (one null); accumulator seeded with
// the bias so D = A*B + C does the add for free.
// Each wave owns one 16x16 tile; 4 waves per block cover 4 N-tiles.
// ---------------------------------------------------------------------------
__global__ __launch_bounds__(128)
void wmma_gemm_nt_kernel(const float* __restrict__ A,
                         const float* __restrict__ W,
                         const float* __restrict__ bias_vec,
                         const float* __restrict__ bias_mat,
                         float* __restrict__ C,
                         int M, int N, int K)
{
    const int wave = threadIdx.x >> 5;
    const int lane = threadIdx.x & 31;
    const int r    = lane & 15;       // row-in-tile / col-in-tile selector
    const int hi   = lane >> 4;       // 0: K={0,1}, 1: K={2,3} (per ISA A/B layout)

    const int m0  = blockIdx.y * 16;
    const int n0  = (blockIdx.x * 4 + wave) * 16;
    const int col = n0 + r;

    // Seed accumulator with bias (C/D layout: VGPR j -> M = j + 8*hi, N = lane&15)
    v8f acc;
    if (bias_mat) {
#pragma unroll
        for (int j = 0; j < 8; ++j)
            acc[j] = bias_mat[(size_t)(m0 + j + 8 * hi) * N + col];
    } else {
        float bv = bias_vec[col];
#pragma unroll
        for (int j = 0; j < 8; ++j)
            acc[j] = bv;
    }

    const float* __restrict__ Arow = A + (size_t)(m0 + r) * K;  // A-frag: row m0+r
    const float* __restrict__ Wrow = W + (size_t)(n0 + r) * K;  // B-frag: Bt[k,n]=W[n,k]

    // Warm the K-streams (global_prefetch_b8); one hint per 128B line.
    for (int p = 0; p < K; p += 32) {
        __builtin_prefetch(Arow + p, 0, 1);
        __builtin_prefetch(Wrow + p, 0, 1);
    }

    // K is always a multiple of 16 here (784, 128) -> unroll x4 is exact and
    // lets the scheduler keep 4 global_load_b64 pairs in flight per WMMA burst.
#pragma unroll 4
    for (int k0 = 0; k0 < K; k0 += 4) {
        v2f a, b;
        a.x = Arow[k0 + 2 * hi];
        a.y = Arow[k0 + 2 * hi + 1];
        b.x = Wrow[k0 + 2 * hi];
        b.y = Wrow[k0 + 2 * hi + 1];
        // v_wmma_f32_16x16x4_f32 : D = A(16x4) * B(4x16) + C
        acc = __builtin_amdgcn_wmma_f32_16x16x4_f32(
            /*neg_a=*/false, a, /*neg_b=*/false, b,
            /*c_mod=*/(short)0, acc, /*reuse_a=*/false, /*reuse_b=*/false);
    }

#pragma unroll
    for (int j = 0; j < 8; ++j)
        C[(size_t)(m0 + j + 8 * hi) * N + col] = acc[j];
}

// ---------------------------------------------------------------------------
// Elementwise tanh
// ---------------------------------------------------------------------------
__global__ __launch_bounds__(256)
void tanh_kernel(const float* __restrict__ src, float* __restrict__ dst, int n)
{
    int e = blockIdx.x * 256 + threadIdx.x;
    if (e < n) dst[e] = tanhf(src[e]);
}

// ---------------------------------------------------------------------------
// g = z - tanh(z_lin); sech2 = 1 - tanh^2; block partial sums of g^2
// ---------------------------------------------------------------------------
__global__ __launch_bounds__(256)
void resid_kernel(const float* __restrict__ znl, const float* __restrict__ zlin,
                  float* __restrict__ g, float* __restrict__ s2,
                  float* __restrict__ partials, int n)
{
    __shared__ float red[256];
    int e = blockIdx.x * 256 + threadIdx.x;
    float p = 0.0f;
    if (e < n) {
        float th = tanhf(zlin[e]);
        float gg = znl[e] - th;
        g[e]  = gg;
        s2[e] = 1.0f - th * th;     // sech^2 = 1 - tanh^2
        p = gg * gg;
    }
    red[threadIdx.x] = p;
    __syncthreads();
#pragma unroll
    for (int s = 128; s > 0; s >>= 1) {
        if (threadIdx.x < s) red[threadIdx.x] += red[threadIdx.x + s];
        __syncthreads();
    }
    if (threadIdx.x == 0) partials[blockIdx.x] = red[0];
}

// Deterministic fixed-order final reduction -> err^2
__global__ __launch_bounds__(256)
void reduce_kernel(const float* __restrict__ partials, int nParts,
                   float* __restrict__ errsq)
{
    __shared__ float red[256];
    float s = 0.0f;
    for (int i = threadIdx.x; i < nParts; i += 256) s += partials[i];
    red[threadIdx.x] = s;
    __syncthreads();
#pragma unroll
    for (int st = 128; st > 0; st >>= 1) {
        if (threadIdx.x < st) red[threadIdx.x] += red[threadIdx.x + st];
        __syncthreads();
    }
    if (threadIdx.x == 0) errsq[0] = red[0];
}

// ---------------------------------------------------------------------------
// Per batch row: async-stream W (64KB) into LDS via the CDNA5 async engine
// (global_load_async_to_lds_b128, ASYNCcnt), transform in place to
// J = I - diag(sech2_row) * W, then unpivoted Gaussian elimination with
// float2 LDS row updates (pitch 130 => banks (2t+j)%64 distinct across the
// 32 lanes, conflict-free), back-substitution, guarded Newton update:
//   z = (err < TOL) ? z : z - dz
// One 128-thread workgroup (4 waves) per row; thread t owns J row t.
// LDS: 128*130*4 + 2*128*4 = 67,584 B  (<= 320 KB/WG; ~4 WGs/WGP).
// ---------------------------------------------------------------------------
#define JP 130
__global__ __launch_bounds__(128)
void solve_update_kernel(const float* __restrict__ W,
                         const float* __restrict__ s2,
                         const float* __restrict__ g,
                         const float* __restrict__ errsq,
                         float* __restrict__ z, float tol2)
{
    __shared__ float Jsh[U_SZ * JP];
    __shared__ float svec[U_SZ];
    __shared__ float rhs[U_SZ];

    const int row = blockIdx.x;
    const int t   = threadIdx.x;   // 0..127

    svec[t] = s2[(size_t)row * U_SZ + t];
    rhs[t]  = g[(size_t)row * U_SZ + t];

    // --- Async copy W[128,128] -> Jsh (pitch JP), 16B chunks per lane. ---
    // 4096 float4 chunks; thread t handles chunks t, t+128, ... (32 each).
    // LDS dest address = low 32 bits of the flat shared pointer (aperture
    // lives in addr[63:32], so truncation yields the LDS byte offset).
    for (int c = t; c < (U_SZ * U_SZ) / 4; c += 128) {
        const int   mrow = c >> 5;            // 32 chunks per 128-float row
        const int   mcol = (c & 31) * 4;
        unsigned    lds  = (unsigned)(uintptr_t)(Jsh + mrow * JP + mcol);
        const float* gp  = W + 4 * c;
        asm volatile("global_load_async_to_lds_b128 %0, %1, off"
                     :: "v"(lds), "v"(gp) : "memory");
    }
    asm volatile("s_wait_asynccnt 0x0" ::: "memory");
    __syncthreads();

    // In-place transform: J = I - diag(svec) * W  (thread t owns column t)
    for (int m = 0; m < U_SZ; ++m) {
        float v = -svec[m] * Jsh[m * JP + t];
        if (m == t) v += 1.0f;
        Jsh[m * JP + t] = v;
    }
    __syncthreads();

    // Forward elimination (unpivoted; J = I - DW well-conditioned here).
    // Row updates vectorized as float2 (b64 LDS ops); js rounded down to even
    // (extra column k of rows t>k is dead after the factor read).
    for (int k = 0; k < U_SZ - 1; ++k) {
        float inv = 1.0f / Jsh[k * JP + k];   // LDS broadcast
        float pr  = rhs[k];
        if (t > k) {
            float f  = Jsh[t * JP + k] * inv;
            const int js = (k + 1) & ~1;
            v2f* __restrict__ Jr = (v2f*)(Jsh + t * JP + js);
            v2f* __restrict__ Pk = (v2f*)(Jsh + k * JP + js);
            const int nj = (U_SZ - js) >> 1;
            for (int jj = 0; jj < nj; ++jj)
                Jr[jj] -= f * Pk[jj];
            rhs[t] -= f * pr;
        }
        __syncthreads();
    }

    // Back substitution: rhs becomes dz
    for (int k = U_SZ - 1; k >= 0; --k) {
        if (t == k) rhs[k] = rhs[k] / Jsh[k * JP + k];
        __syncthreads();
        if (t < k) rhs[t] -= Jsh[t * JP + k] * rhs[k];
        __syncthreads();
    }

    // Guarded Newton update (emulates jnp.where(err < TOL, z, z - dz))
    bool  conv = (errsq[0] < tol2);
    float zi   = z[(size_t)row * U_SZ + t];
    z[(size_t)row * U_SZ + t] = conv ? zi : (zi - rhs[t]);
}

// ---------------------------------------------------------------------------
// logits = z @ Wo^T + bo ; softmax.  One wave32 per batch row.
// ---------------------------------------------------------------------------
__global__ __launch_bounds__(32)
void out_softmax_kernel(const float* __restrict__ z,
                        const float* __restrict__ Wo,
                        const float* __restrict__ bo,
                        float* __restrict__ out)
{
    const int row  = blockIdx.x;
    const int lane = threadIdx.x;
    const float* zr = z + (size_t)row * U_SZ;

    float logits[DOUT_SZ];
#pragma unroll
    for (int o = 0; o < DOUT_SZ; ++o) {
        const float* w = Wo + o * U_SZ;
        float s = 0.0f;
        for (int k = lane; k < U_SZ; k += 32) s += zr[k] * w[k];
#pragma unroll
        for (int off = 16; off > 0; off >>= 1) s += __shfl_xor(s, off, 32);
        logits[o] = s + bo[o];
    }
    float m = logits[0];
#pragma unroll
    for (int o = 1; o < DOUT_SZ; ++o) m = fmaxf(m, logits[o]);
    float den = 0.0f;
#pragma unroll
    for (int o = 0; o < DOUT_SZ; ++o) den += expf(logits[o] - m);
    float inv = 1.0f / den;
#pragma unroll
    for (int o = 0; o < DOUT_SZ; ++o)
        if (lane == o) out[(size_t)row * DOUT_SZ + o] = expf(logits[o] - m) * inv;
}

// ---------------------------------------------------------------------------
extern "C" void kernel_launch(void* const* d_in, const int* in_sizes, int n_in,
                              void* d_out, int out_size, void* d_ws, size_t ws_size,
                              hipStream_t stream)
{
    (void)in_sizes; (void)n_in; (void)out_size; (void)ws_size;
    const float* x  = (const float*)d_in[0];   // [2048,784]
    const float* Wi = (const float*)d_in[1];   // [128,784]
    const float* bi = (const float*)d_in[2];   // [128]
    const float* Wb = (const float*)d_in[3];   // [2,128,128]
    const float* Wo = (const float*)d_in[4];   // [10,128]
    const float* bo = (const float*)d_in[5];   // [10]
    float* out = (float*)d_out;                // [2048,10]

    const int nelem = B_SZ * U_SZ;             // 262144
    const int eb    = nelem / 256;             // 1024 blocks
    const float TOL2 = 1e-4f * 1e-4f;

    float* ws    = (float*)d_ws;
    float* xbuf  = ws + 0 * (size_t)nelem;     // layer input ("x" inside Newton)
    float* znl   = ws + 1 * (size_t)nelem;     // Newton iterate z
    float* zlin  = ws + 2 * (size_t)nelem;     // z@W^T + x
    float* gbuf  = ws + 3 * (size_t)nelem;     // residual
    float* s2    = ws + 4 * (size_t)nelem;     // sech^2
    float* parts = ws + 5 * (size_t)nelem;     // [1024] block partials
    float* errsq = parts + 1024;               // scalar err^2

    dim3 gemmGrid(U_SZ / 64, B_SZ / 16);       // 4 waves/block * 16 cols = 64 cols

    // z_in = x @ Wi^T + bi
    wmma_gemm_nt_kernel<<<gemmGrid, 128, 0, stream>>>(
        x, Wi, bi, nullptr, xbuf, B_SZ, U_SZ, DIN_SZ);

    for (int l = 0; l < 2; ++l) {
        const float* Wl = Wb + (size_t)l * U_SZ * U_SZ;
        // z0 = tanh(x)
        tanh_kernel<<<eb, 256, 0, stream>>>(xbuf, znl, nelem);
        for (int it = 0; it < MAXIT; ++it) {
            wmma_gemm_nt_kernel<<<gemmGrid, 128, 0, stream>>>(
                znl, Wl, nullptr, xbuf, zlin, B_SZ, U_SZ, U_SZ);
            resid_kernel<<<eb, 256, 0, stream>>>(znl, zlin, gbuf, s2, parts, nelem);
            reduce_kernel<<<1, 256, 0, stream>>>(parts, eb, errsq);
            solve_update_kernel<<<B_SZ, 128, 0, stream>>>(Wl, s2, gbuf, errsq, znl, TOL2);
        }
        // out = tanh(z* @ W^T + x) becomes next layer's input
        wmma_gemm_nt_kernel<<<gemmGrid, 128, 0, stream>>>(
            znl, Wl, nullptr, xbuf, zlin, B_SZ, U_SZ, U_SZ);
        tanh_kernel<<<eb, 256, 0, stream>>>(zlin, xbuf, nelem);
    }

    out_softmax_kernel<<<B_SZ, 32, 0, stream>>>(xbuf, Wo, bo, out);
}